// DREMLayer_56831007261107
// MI455X (gfx1250) — compile-verified
//
#include <hip/hip_runtime.h>
#include <hip/hip_bf16.h>

// ---------------------------------------------------------------------------
// DREM relational-attention layer for MI455X (gfx1250, wave32, WMMA).
// All matmuls use V_WMMA_F32_16X16X4_F32 (fp32 is required: attention logits
// are O(1e4..1e5) because adj_scale ~ 5e3, so softmax is near-one-hot and
// low-precision logits would change the result).
// Each wave computes a 32x32 output tile (2x2 WMMA accumulators): 4 WMMAs per
// k-step on 4 operand loads -> 2x operand reuse vs a 16x16 tile.
// ---------------------------------------------------------------------------

typedef __attribute__((ext_vector_type(2))) float v2f;
typedef __attribute__((ext_vector_type(8))) float v8f;

#define BB   4
#define NN   2048
#define DIN  256
#define DOUT 256
#define HH   8
#define HD   32
#define RR   5
#define KBH  (BB * HD)          // 128 : contraction dim for scores / width of per-head h
#define MROWS (BB * NN)         // 8192

__device__ __forceinline__ v8f wmma4(v2f a, v2f b, v8f c) {
    // D = A(16x4,f32) * B(4x16,f32) + C(16x16,f32)
    return __builtin_amdgcn_wmma_f32_16x16x4_f32(
        /*neg_a=*/false, a, /*neg_b=*/false, b,
        /*c_mod=*/(short)0, c, /*reuse_a=*/false, /*reuse_b=*/false);
}

#define ZERO8 {0.f, 0.f, 0.f, 0.f, 0.f, 0.f, 0.f, 0.f}

// ---------------------------------------------------------------------------
// 1) adj_scale[b,n] = sum_{r,m} adj[b,r,n,m]      (bandwidth-bound, 335 MB)
// ---------------------------------------------------------------------------
__global__ void adj_reduce_kernel(const float* __restrict__ adj,
                                  float* __restrict__ adj_scale) {
    const int bn = blockIdx.x;                  // 0 .. B*N-1
    const int b  = bn >> 11;
    const int n  = bn & (NN - 1);
    const int t  = threadIdx.x;                 // 256 threads

    float acc = 0.0f;
    for (int r = 0; r < RR; ++r) {
        const float4* row4 =
            (const float4*)(adj + ((size_t)(b * RR + r) * NN + n) * NN);
        #pragma unroll
        for (int m4 = t; m4 < NN / 4; m4 += 256) {
            __builtin_prefetch(&row4[m4 + 256], 0, 1);   // global_prefetch_b8
            float4 v = row4[m4];
            acc += (v.x + v.y) + (v.z + v.w);
        }
    }
    __shared__ float red[256];
    red[t] = acc;
    __syncthreads();
    for (int s = 128; s > 0; s >>= 1) {
        if (t < s) red[t] += red[t + s];
        __syncthreads();
    }
    if (t == 0) adj_scale[bn] = red[0];
}

// ---------------------------------------------------------------------------
// 2) Wr_sum = sum_r Wr_w[r]   (H*HD*DIN = 65536)  ;  br_sum = sum_r Wr_b[r]
// ---------------------------------------------------------------------------
__global__ void wr_reduce_kernel(const float* __restrict__ Wr_w,
                                 const float* __restrict__ Wr_b,
                                 float* __restrict__ wr_sum,
                                 float* __restrict__ br_sum) {
    const int idx = blockIdx.x * 256 + threadIdx.x;     // < 65536
    float s = 0.0f;
    #pragma unroll
    for (int r = 0; r < RR; ++r) s += Wr_w[r * (HH * HD * DIN) + idx];
    wr_sum[idx] = s;
    if (idx < HH * HD) {
        float sb = 0.0f;
        #pragma unroll
        for (int r = 0; r < RR; ++r) sb += Wr_b[r * (HH * HD) + idx];
        br_sum[idx] = sb;
    }
}

// ---------------------------------------------------------------------------
// 3) Projection GEMM (32x32 tile/wave): rows m = b*N+n over 8192,
//    cols j = head*32+d over 256.
//    USE_SCALE==0 :  h = x @ W^T + bias            -> hbuf[head][n][b*32+d]
//    USE_SCALE==1 :  r = scale[m]*(x @ W^T) + bias -> rbuf[head][n][b*32+d]
// ---------------------------------------------------------------------------
template <int USE_SCALE>
__global__ void proj_gemm_kernel(const float* __restrict__ x,     // [8192][256]
                                 const float* __restrict__ W,     // [256][256]
                                 const float* __restrict__ bias,  // [256]
                                 const float* __restrict__ scale, // [8192]
                                 float* __restrict__ outbuf) {    // [8][2048][128]
    const int lane = threadIdx.x & 31;
    const int half = lane >> 4;          // 0: K pair {0,1}, 1: K pair {2,3}
    const int rc   = lane & 15;          // M (for A) / N (for B,C,D) within 16-tile
    const int m0 = blockIdx.x * 32;
    const int j0 = blockIdx.y * 32;

    v8f c00 = ZERO8, c01 = ZERO8, c10 = ZERO8, c11 = ZERO8;
    #pragma unroll 4
    for (int kk = 0; kk < DIN; kk += 4) {
        const int k = kk + 2 * half;
        v2f a0 = *(const v2f*)&x[(size_t)(m0 + rc) * DIN + k];
        v2f a1 = *(const v2f*)&x[(size_t)(m0 + 16 + rc) * DIN + k];
        v2f b0 = *(const v2f*)&W[(size_t)(j0 + rc) * DIN + k];
        v2f b1 = *(const v2f*)&W[(size_t)(j0 + 16 + rc) * DIN + k];
        c00 = wmma4(a0, b0, c00);
        c01 = wmma4(a0, b1, c01);
        c10 = wmma4(a1, b0, c10);
        c11 = wmma4(a1, b1, c11);
    }

    #pragma unroll
    for (int mt = 0; mt < 2; ++mt) {
        #pragma unroll
        for (int nt = 0; nt < 2; ++nt) {
            const v8f& c = (mt == 0) ? (nt == 0 ? c00 : c01)
                                     : (nt == 0 ? c10 : c11);
            const int j    = j0 + nt * 16 + rc;
            const int head = j >> 5;
            const int d    = j & 31;
            const float bj = bias[j];
            #pragma unroll
            for (int i = 0; i < 8; ++i) {
                const int m  = m0 + mt * 16 + i + 8 * half;
                const int b_ = m >> 11;
                const int n  = m & (NN - 1);
                float v = c[i];
                if (USE_SCALE) v = scale[m] * v + bj; else v = v + bj;
                outbuf[(size_t)head * NN * KBH + (size_t)n * KBH + b_ * HD + d] = v;
            }
        }
    }
}

// ---------------------------------------------------------------------------
// 4) Per-head scores GEMM (32x32 tile/wave):
//    scores[n][m] = leaky( hA[n][:] . rB[m][:] ),  K = 128
// ---------------------------------------------------------------------------
__global__ void scores_gemm_kernel(const float* __restrict__ hA,  // [2048][128]
                                   const float* __restrict__ rB,  // [2048][128]
                                   float* __restrict__ scores) {  // [2048][2048]
    const int lane = threadIdx.x & 31;
    const int half = lane >> 4;
    const int rc   = lane & 15;
    const int n0 = blockIdx.x * 32;
    const int m0 = blockIdx.y * 32;

    v8f c00 = ZERO8, c01 = ZERO8, c10 = ZERO8, c11 = ZERO8;
    #pragma unroll 4
    for (int kk = 0; kk < KBH; kk += 4) {
        const int k = kk + 2 * half;
        v2f a0 = *(const v2f*)&hA[(size_t)(n0 + rc) * KBH + k];
        v2f a1 = *(const v2f*)&hA[(size_t)(n0 + 16 + rc) * KBH + k];
        v2f b0 = *(const v2f*)&rB[(size_t)(m0 + rc) * KBH + k];
        v2f b1 = *(const v2f*)&rB[(size_t)(m0 + 16 + rc) * KBH + k];
        c00 = wmma4(a0, b0, c00);
        c01 = wmma4(a0, b1, c01);
        c10 = wmma4(a1, b0, c10);
        c11 = wmma4(a1, b1, c11);
    }
    #pragma unroll
    for (int mt = 0; mt < 2; ++mt) {
        #pragma unroll
        for (int nt = 0; nt < 2; ++nt) {
            const v8f& c = (mt == 0) ? (nt == 0 ? c00 : c01)
                                     : (nt == 0 ? c10 : c11);
            #pragma unroll
            for (int i = 0; i < 8; ++i) {
                const int n = n0 + mt * 16 + i + 8 * half;
                float s = c[i];
                s = (s >= 0.0f) ? s : 0.2f * s;          // leaky ReLU, slope 0.2
                scores[(size_t)n * NN + (m0 + nt * 16 + rc)] = s;
            }
        }
    }
}

// ---------------------------------------------------------------------------
// 5) Row softmax over 2048 columns (one block per row).
// ---------------------------------------------------------------------------
__global__ void softmax_kernel(float* __restrict__ scores) {
    float* rowp = scores + (size_t)blockIdx.x * NN;
    const int t = threadIdx.x;                           // 256 threads
    __shared__ float red[256];

    float mx = -INFINITY;
    for (int m = t; m < NN; m += 256) mx = fmaxf(mx, rowp[m]);
    red[t] = mx;
    __syncthreads();
    for (int s = 128; s > 0; s >>= 1) {
        if (t < s) red[t] = fmaxf(red[t], red[t + s]);
        __syncthreads();
    }
    mx = red[0];
    __syncthreads();

    float sum = 0.0f;
    for (int m = t; m < NN; m += 256) {
        float e = expf(rowp[m] - mx);
        rowp[m] = e;
        sum += e;
    }
    red[t] = sum;
    __syncthreads();
    for (int s = 128; s > 0; s >>= 1) {
        if (t < s) red[t] += red[t + s];
        __syncthreads();
    }
    const float inv = 1.0f / red[0];
    for (int m = t; m < NN; m += 256) rowp[m] *= inv;
}

// ---------------------------------------------------------------------------
// 6) Per-head attention GEMM (32x32 tile/wave):
//    oh[n][c] = sum_m attn[n][m] * hB[m][c],  K = 2048
// ---------------------------------------------------------------------------
__global__ void attn_gemm_kernel(const float* __restrict__ attn, // [2048][2048]
                                 const float* __restrict__ hB,   // [2048][128]
                                 float* __restrict__ oh) {       // [2048][128]
    const int lane = threadIdx.x & 31;
    const int half = lane >> 4;
    const int rc   = lane & 15;
    const int n0 = blockIdx.x * 32;
    const int c0 = blockIdx.y * 32;

    v8f c00 = ZERO8, c01 = ZERO8, c10 = ZERO8, c11 = ZERO8;
    #pragma unroll 4
    for (int kk = 0; kk < NN; kk += 4) {
        const int k = kk + 2 * half;
        v2f a0 = *(const v2f*)&attn[(size_t)(n0 + rc) * NN + k];
        v2f a1 = *(const v2f*)&attn[(size_t)(n0 + 16 + rc) * NN + k];
        v2f b0, b1;
        b0.x = hB[(size_t)(k + 0) * KBH + (c0 + rc)];
        b0.y = hB[(size_t)(k + 1) * KBH + (c0 + rc)];
        b1.x = hB[(size_t)(k + 0) * KBH + (c0 + 16 + rc)];
        b1.y = hB[(size_t)(k + 1) * KBH + (c0 + 16 + rc)];
        c00 = wmma4(a0, b0, c00);
        c01 = wmma4(a0, b1, c01);
        c10 = wmma4(a1, b0, c10);
        c11 = wmma4(a1, b1, c11);
    }
    #pragma unroll
    for (int mt = 0; mt < 2; ++mt) {
        #pragma unroll
        for (int nt = 0; nt < 2; ++nt) {
            const v8f& c = (mt == 0) ? (nt == 0 ? c00 : c01)
                                     : (nt == 0 ? c10 : c11);
            #pragma unroll
            for (int i = 0; i < 8; ++i) {
                const int n = n0 + mt * 16 + i + 8 * half;
                oh[(size_t)n * KBH + (c0 + nt * 16 + rc)] = c[i];
            }
        }
    }
}

// ---------------------------------------------------------------------------
// 7) Final GEMM (32x32 tile/wave) + bias + relu(leaky(.)) == relu(.):
//    out[m][o] = relu( sum_k ohbuf[k>>5][n][b*32+(k&31)] * out_w[o][k] + out_b[o] )
// ---------------------------------------------------------------------------
__global__ void final_gemm_kernel(const float* __restrict__ ohbuf, // [8][2048][128]
                                  const float* __restrict__ out_w, // [256][256]
                                  const float* __restrict__ out_b, // [256]
                                  float* __restrict__ out) {       // [8192][256]
    const int lane = threadIdx.x & 31;
    const int half = lane >> 4;
    const int rc   = lane & 15;
    const int m0 = blockIdx.x * 32;
    const int o0 = blockIdx.y * 32;

    const int mA0 = m0 + rc,       bA0 = mA0 >> 11, nA0 = mA0 & (NN - 1);
    const int mA1 = m0 + 16 + rc,  bA1 = mA1 >> 11, nA1 = mA1 & (NN - 1);

    v8f c00 = ZERO8, c01 = ZERO8, c10 = ZERO8, c11 = ZERO8;
    #pragma unroll 4
    for (int kk = 0; kk < HH * HD; kk += 4) {
        const int k    = kk + 2 * half;          // k, k+1 stay in the same head
        const int head = k >> 5;
        const int d    = k & 31;
        const size_t hb = (size_t)head * NN * KBH + d;
        v2f a0 = *(const v2f*)&ohbuf[hb + (size_t)nA0 * KBH + bA0 * HD];
        v2f a1 = *(const v2f*)&ohbuf[hb + (size_t)nA1 * KBH + bA1 * HD];
        v2f b0 = *(const v2f*)&out_w[(size_t)(o0 + rc) * (HH * HD) + k];
        v2f b1 = *(const v2f*)&out_w[(size_t)(o0 + 16 + rc) * (HH * HD) + k];
        c00 = wmma4(a0, b0, c00);
        c01 = wmma4(a0, b1, c01);
        c10 = wmma4(a1, b0, c10);
        c11 = wmma4(a1, b1, c11);
    }
    #pragma unroll
    for (int mt = 0; mt < 2; ++mt) {
        #pragma unroll
        for (int nt = 0; nt < 2; ++nt) {
            const v8f& c = (mt == 0) ? (nt == 0 ? c00 : c01)
                                     : (nt == 0 ? c10 : c11);
            const int o    = o0 + nt * 16 + rc;
            const float bo = out_b[o];
            #pragma unroll
            for (int i = 0; i < 8; ++i) {
                const int mi = m0 + mt * 16 + i + 8 * half;
                float v = c[i] + bo;
                out[(size_t)mi * DOUT + o] = fmaxf(v, 0.0f); // relu∘leaky == relu
            }
        }
    }
}

// ---------------------------------------------------------------------------
// Launcher
// ---------------------------------------------------------------------------
extern "C" void kernel_launch(void* const* d_in, const int* in_sizes, int n_in,
                              void* d_out, int out_size, void* d_ws, size_t ws_size,
                              hipStream_t stream) {
    (void)in_sizes; (void)n_in; (void)out_size; (void)ws_size;

    const float* x     = (const float*)d_in[0];  // [4,2048,256]
    const float* adj   = (const float*)d_in[1];  // [4,5,2048,2048]
    const float* W_w   = (const float*)d_in[2];  // [8,32,256] -> [256][256]
    const float* W_b   = (const float*)d_in[3];  // [256]
    const float* Wr_w  = (const float*)d_in[4];  // [5,8,32,256]
    const float* Wr_b  = (const float*)d_in[5];  // [5,256]
    const float* out_w = (const float*)d_in[6];  // [256,256]
    const float* out_b = (const float*)d_in[7];  // [256]
    float*       out   = (float*)d_out;          // [4,2048,256]

    // Workspace layout (floats), total ~10.56M floats (~42 MB)
    float* ws        = (float*)d_ws;
    float* adj_scale = ws;                                   // 8192
    float* wr_sum    = adj_scale + MROWS;                    // 65536
    float* br_sum    = wr_sum    + HH * HD * DIN;            // 256
    float* hbuf      = br_sum    + HH * HD;                  // 8*2048*128
    float* rbuf      = hbuf      + (size_t)HH * NN * KBH;    // 8*2048*128
    float* ohbuf     = rbuf      + (size_t)HH * NN * KBH;    // 8*2048*128
    float* scores    = ohbuf     + (size_t)HH * NN * KBH;    // 2048*2048 (per-head reuse)

    // 1) adj reduction (bandwidth-bound over 335 MB)
    adj_reduce_kernel<<<MROWS, 256, 0, stream>>>(adj, adj_scale);

    // 2) relation-summed weights
    wr_reduce_kernel<<<(HH * HD * DIN) / 256, 256, 0, stream>>>(Wr_w, Wr_b,
                                                                wr_sum, br_sum);

    // 3) projections h and r (WMMA GEMM, 32x32 tile per wave)
    {
        dim3 grid(MROWS / 32, (HH * HD) / 32);
        proj_gemm_kernel<0><<<grid, 32, 0, stream>>>(x, W_w, W_b, adj_scale, hbuf);
        proj_gemm_kernel<1><<<grid, 32, 0, stream>>>(x, wr_sum, br_sum, adj_scale,
                                                     rbuf);
    }

    // 4-6) per-head: scores GEMM -> softmax -> attention GEMM (scores reused)
    for (int h = 0; h < HH; ++h) {
        const float* hA = hbuf  + (size_t)h * NN * KBH;
        const float* rB = rbuf  + (size_t)h * NN * KBH;
        float*       oh = ohbuf + (size_t)h * NN * KBH;

        dim3 gs(NN / 32, NN / 32);
        scores_gemm_kernel<<<gs, 32, 0, stream>>>(hA, rB, scores);

        softmax_kernel<<<NN, 256, 0, stream>>>(scores);

        dim3 ga(NN / 32, KBH / 32);
        attn_gemm_kernel<<<ga, 32, 0, stream>>>(scores, hA, oh);
    }

    // 7) output projection + relu
    {
        dim3 grid(MROWS / 32, DOUT / 32);
        final_gemm_kernel<<<grid, 32, 0, stream>>>(ohbuf, out_w, out_b, out);
    }
}